// CoxSquareLoss_52922587021938
// MI455X (gfx1250) — compile-verified
//
#include <hip/hip_runtime.h>
#include <hip/hip_bf16.h>
#include <math.h>
#include <stdint.h>

typedef __attribute__((ext_vector_type(2))) float v2f;
typedef __attribute__((ext_vector_type(8))) float v8f;

#define T_POOL (1u << 20)
#define T_MAX  (T_POOL - 1u)
#define N_TILES 4096  // T_POOL / 256

// ---------------- workspace layout ----------------
// [0, 4MiB)          : Arev  float[T_POOL]   exp-sum histogram, reverse-indexed
// [4MiB, 8MiB)       : Drev  uint [T_POOL]   event-count histogram, reverse-indexed
// [8MiB, +16KiB)     : Tsum  float[4096]     per-256-bucket tile sums (scan order)
// [+16KiB, +32KiB)   : TP    float[4096]     inclusive scan of Tsum
// [+32KiB, +32KiB+16): Acc   double[2]       {sum(ev*x), sum(D*log S)}

__global__ void zero_ws(uint4* ws, int n16) {
  int i = blockIdx.x * blockDim.x + threadIdx.x;
  if (i < n16) ws[i] = make_uint4(0u, 0u, 0u, 0u);
}

__device__ __forceinline__ float readlane_f(float v, int srcLane) {
  return __int_as_float(__builtin_amdgcn_readlane(__float_as_int(v), srcLane));
}

// ---------------------------------------------------------------------------
// 256-element inclusive prefix scan within one wave32 using
// V_WMMA_F32_16X16X4_F32 and triangular ones-matrices.
//   X[r][c] = src[16r + c]
//   step 1: C1 = sum_g X[:,4g:4g+4] * U_g        (U[k][c] = k<=c)  row prefixes
//   step 2: C  = C1 + sum_g Ls[:,4g:4g+4] * T_g  (Ls[r][k]=k<r, T[k][*]=rowtot[k])
// Lane l holds, in component k, the inclusive prefix at linear index
//         16*(k + 8*(l>>4)) + (l&15).
// Row totals are wave-uniform -> v_readlane into SGPRs (no LDS traffic).
// ---------------------------------------------------------------------------
__device__ __forceinline__ v8f wmma_scan256(const float* __restrict__ src, int lane) {
  const int hi  = lane >> 4;
  const int col = lane & 15;
  v8f c = {0.f, 0.f, 0.f, 0.f, 0.f, 0.f, 0.f, 0.f};

  // A-format slabs of X: lane l supplies row (l&15), K = 4g + 2*hi + {0,1}
  v2f a[4];
#pragma unroll
  for (int g = 0; g < 4; ++g) {
    const int k0 = 4 * g + 2 * hi;
    v2f av;
    av.x = src[16 * col + k0];
    av.y = src[16 * col + k0 + 1];
    a[g] = av;
  }
#pragma unroll
  for (int g = 0; g < 4; ++g) {
    const int k0 = 4 * g + 2 * hi;
    v2f b;
    b.x = (k0     <= col) ? 1.f : 0.f;
    b.y = (k0 + 1 <= col) ? 1.f : 0.f;
    c = __builtin_amdgcn_wmma_f32_16x16x4_f32(false, a[g], false, b, (short)0, c,
                                              false, false);
  }
  // Broadcast row totals (column 15 of C1): rows 0..7 in lane 15, 8..15 in 31.
  float rt[16];
#pragma unroll
  for (int k = 0; k < 8; ++k) {
    rt[k]     = readlane_f(c[k], 15);
    rt[k + 8] = readlane_f(c[k], 31);
  }
#pragma unroll
  for (int g = 0; g < 4; ++g) {
    const int k0 = 4 * g + 2 * hi;
    v2f la;
    la.x = (k0     < col) ? 1.f : 0.f;
    la.y = (k0 + 1 < col) ? 1.f : 0.f;
    v2f tb;
    tb.x = hi ? rt[4 * g + 2] : rt[4 * g + 0];
    tb.y = hi ? rt[4 * g + 3] : rt[4 * g + 1];
    c = __builtin_amdgcn_wmma_f32_16x16x4_f32(false, la, false, tb, (short)0, c,
                                              false, false);
  }
  return c;
}

// ---------------------------------------------------------------------------
// Pass 1: streaming histogram, double-buffered async-to-LDS pipeline.
// Chunk i+1 is issued with global_load_async_to_lds_b128 while chunk i is
// consumed; s_wait_asynccnt 3 releases the current buffer (per-wave async
// loads complete in order) while the next 3 stay in flight.
// ---------------------------------------------------------------------------
__global__ void __launch_bounds__(256) histo_kernel(
    const float* __restrict__ x, const int* __restrict__ times,
    const int* __restrict__ events, float* __restrict__ Arev,
    unsigned* __restrict__ Drev, double* __restrict__ Acc, int nPerBlock) {
  __shared__ __align__(16) float sx[2][1024];
  __shared__ __align__(16) int   st[2][1024];
  __shared__ __align__(16) int   se[2][1024];
  __shared__ double red[8];

  const int tid = threadIdx.x;
  const uint32_t lx0 = (uint32_t)(uintptr_t)&sx[0][0] + (uint32_t)tid * 16u;
  const uint32_t lt0 = (uint32_t)(uintptr_t)&st[0][0] + (uint32_t)tid * 16u;
  const uint32_t le0 = (uint32_t)(uintptr_t)&se[0][0] + (uint32_t)tid * 16u;
  const uint64_t gx = (uint64_t)(uintptr_t)x;
  const uint64_t gt = (uint64_t)(uintptr_t)times;
  const uint64_t ge = (uint64_t)(uintptr_t)events;

  auto issue = [&](int chunk, int buf) {
    const uint32_t voff =
        ((unsigned)blockIdx.x * (unsigned)nPerBlock + ((unsigned)chunk << 10) +
         ((unsigned)tid << 2)) << 2;  // byte offset: 4 records * 4B per lane
    const uint32_t bofs = (unsigned)buf << 12;  // 4096B per buffer
    asm volatile("global_load_async_to_lds_b128 %0, %1, %2 offset:0"
                 :: "v"(lx0 + bofs), "v"(voff), "s"(gx) : "memory");
    asm volatile("global_load_async_to_lds_b128 %0, %1, %2 offset:0"
                 :: "v"(lt0 + bofs), "v"(voff), "s"(gt) : "memory");
    asm volatile("global_load_async_to_lds_b128 %0, %1, %2 offset:0"
                 :: "v"(le0 + bofs), "v"(voff), "s"(ge) : "memory");
  };

  float evx = 0.f;
  const int iters = nPerBlock >> 10;  // 1024 records per chunk
  issue(0, 0);
  for (int it = 0; it < iters; ++it) {
    const int cur = it & 1;
    if (it + 1 < iters) {
      issue(it + 1, cur ^ 1);
      asm volatile("s_wait_asynccnt 0x3" ::: "memory");  // current buffer done
    } else {
      asm volatile("s_wait_asynccnt 0x0" ::: "memory");
    }
    __syncthreads();
#pragma unroll
    for (int r = 0; r < 4; ++r) {
      const int i = (tid << 2) + r;
      const float xv = sx[cur][i];
      const unsigned ri = T_MAX - (unsigned)st[cur][i];  // reverse index
      unsafeAtomicAdd(&Arev[ri], __expf(xv));
      if (se[cur][i]) {
        atomicAdd(&Drev[ri], 1u);
        evx += xv;
      }
    }
    __syncthreads();  // buffer `cur` free for re-issue next iteration
  }
  // block-reduce sum(ev*x) -> one f64 atomic per block
  double d = (double)evx;
#pragma unroll
  for (int s = 16; s > 0; s >>= 1) d += __shfl_down(d, s, 32);
  if ((tid & 31) == 0) red[tid >> 5] = d;
  __syncthreads();
  if (tid == 0) {
    double s = 0.0;
#pragma unroll
    for (int j = 0; j < 8; ++j) s += red[j];
    unsafeAtomicAdd(&Acc[0], s);
  }
}

// Pass 2a: 256-bucket tile sums (scan order == reverse time order already).
__global__ void __launch_bounds__(256) tile_sum_kernel(
    const float* __restrict__ Arev, float* __restrict__ Tsum) {
  __shared__ float wsum[8];
  const int tid = threadIdx.x;
  float v = Arev[(blockIdx.x << 8) + tid];
#pragma unroll
  for (int s = 16; s > 0; s >>= 1) v += __shfl_down(v, s, 32);
  if ((tid & 31) == 0) wsum[tid >> 5] = v;
  __syncthreads();
  if (tid == 0) {
    float s = 0.f;
#pragma unroll
    for (int j = 0; j < 8; ++j) s += wsum[j];
    Tsum[blockIdx.x] = s;
  }
}

// Pass 2b: inclusive scan of the 4096 tile sums. 16 waves, each WMMA-scans
// 256 values; cross-wave offsets via LDS.
__global__ void __launch_bounds__(512) scan_tile_kernel(
    const float* __restrict__ Tsum, float* __restrict__ TP) {
  __shared__ float wtot[16];
  const int tid = threadIdx.x;
  const int w = tid >> 5, lane = tid & 31;
  v8f p = wmma_scan256(Tsum + (w << 8), lane);
  const float wt = readlane_f(p[7], 31);  // element (15,15) = chunk total
  if (lane == 0) wtot[w] = wt;
  __syncthreads();
  float off = 0.f;
  for (int j = 0; j < 16; ++j) off += (j < w) ? wtot[j] : 0.f;
  const int hi = lane >> 4, col = lane & 15;
#pragma unroll
  for (int k = 0; k < 8; ++k)
    TP[(w << 8) + 16 * (k + 8 * hi) + col] = off + p[k];
}

// Pass 3: per-tile WMMA scan -> S(t) (suffix sums, via reverse indexing),
// accumulate sum_t D[t]*log(S) into f64.
__global__ void __launch_bounds__(32) finalize_kernel(
    const float* __restrict__ Arev, const unsigned* __restrict__ Drev,
    const float* __restrict__ TP, double* __restrict__ Acc) {
  const int tile = blockIdx.x;
  const int lane = threadIdx.x;
  const int hi = lane >> 4, col = lane & 15;
  const float off = tile ? TP[tile - 1] : 0.f;  // exclusive tile offset, exact
  v8f p = wmma_scan256(Arev + (tile << 8), lane);
  double acc = 0.0;
#pragma unroll
  for (int k = 0; k < 8; ++k) {
    const int idx = (tile << 8) + 16 * (k + 8 * hi) + col;
    const unsigned d = Drev[idx];
    if (d) acc += (double)d * (double)__logf(off + p[k]);  // d>0 => S>0
  }
#pragma unroll
  for (int s = 16; s > 0; s >>= 1) acc += __shfl_down(acc, s, 32);
  if (lane == 0) unsafeAtomicAdd(&Acc[1], acc);
}

__global__ void final_kernel(const double* __restrict__ Acc,
                             float* __restrict__ out, int n) {
  const double loss = (Acc[1] - Acc[0]) / (double)n;
  out[0] = (float)sqrt(loss > 0.0 ? loss : 0.0);
}

extern "C" void kernel_launch(void* const* d_in, const int* in_sizes, int n_in,
                              void* d_out, int out_size, void* d_ws, size_t ws_size,
                              hipStream_t stream) {
  const float* x      = (const float*)d_in[0];
  const int*   times  = (const int*)d_in[1];
  const int*   events = (const int*)d_in[2];
  const int n = in_sizes[0];  // 2^24

  char* ws = (char*)d_ws;
  float*    Arev = (float*)ws;
  unsigned* Drev = (unsigned*)(ws + (size_t)(4u << 20));
  float*    Tsum = (float*)(ws + (size_t)(8u << 20));
  float*    TP   = (float*)(ws + (size_t)(8u << 20) + 16384);
  double*   Acc  = (double*)(ws + (size_t)(8u << 20) + 32768);

  const int zero_bytes = (8 << 20) + 32768 + 16;
  const int n16 = zero_bytes / 16;
  zero_ws<<<(n16 + 255) / 256, 256, 0, stream>>>((uint4*)d_ws, n16);

  const int histBlocks = 2048;
  const int perBlock = n / histBlocks;  // 8192 records per block
  histo_kernel<<<histBlocks, 256, 0, stream>>>(x, times, events, Arev, Drev,
                                               Acc, perBlock);
  tile_sum_kernel<<<N_TILES, 256, 0, stream>>>(Arev, Tsum);
  scan_tile_kernel<<<1, 512, 0, stream>>>(Tsum, TP);
  finalize_kernel<<<N_TILES, 32, 0, stream>>>(Arev, Drev, TP, Acc);
  final_kernel<<<1, 1, 0, stream>>>(Acc, (float*)d_out, n);
}